// SelfAttentionPooling_75376676045154
// MI455X (gfx1250) — compile-verified
//
#include <hip/hip_runtime.h>
#include <hip/hip_bf16.h>

#define S_DIM 1024
#define V_DIM 1024
#define F_DIM 128
#define K_DIM 512

typedef __attribute__((ext_vector_type(2))) float v2f;
typedef __attribute__((ext_vector_type(8))) float v8f;

// ---------------------------------------------------------------------------
// 1) degree of (A + I): dinv[u] = rsqrt(1 + sum_v adj[u,v])
// ---------------------------------------------------------------------------
__global__ void deg_kernel(const float* __restrict__ adj, float* __restrict__ dinv) {
    __shared__ float red[256];
    const int u = blockIdx.x;
    float s = 0.f;
    for (int v = threadIdx.x; v < V_DIM; v += 256) s += adj[u * V_DIM + v];
    red[threadIdx.x] = s;
    __syncthreads();
    for (int off = 128; off > 0; off >>= 1) {
        if (threadIdx.x < off) red[threadIdx.x] += red[threadIdx.x + off];
        __syncthreads();
    }
    if (threadIdx.x == 0) dinv[u] = rsqrtf(red[0] + 1.0f);
}

// ---------------------------------------------------------------------------
// 2) An[u,v] = dinv[u] * (adj[u,v] + (u==v)) * dinv[v]
// ---------------------------------------------------------------------------
__global__ void an_kernel(const float* __restrict__ adj, const float* __restrict__ dinv,
                          float* __restrict__ An) {
    const int i = blockIdx.x * blockDim.x + threadIdx.x;   // V*V = 1M
    const int u = i >> 10;
    const int v = i & (V_DIM - 1);
    const float a = adj[i] + (u == v ? 1.0f : 0.0f);
    An[i] = dinv[u] * a * dinv[v];
}

// ---------------------------------------------------------------------------
// 3) support[row] = dot(x[row, 0:128], W)  -- one wave32 per row, float4 lanes
// ---------------------------------------------------------------------------
__global__ void support_kernel(const float* __restrict__ x, const float* __restrict__ W,
                               float* __restrict__ support) {
    __shared__ float Ws[F_DIM];
    const int t = threadIdx.x;
    if (t < F_DIM) Ws[t] = W[t];
    __syncthreads();
    const int wave = t >> 5;
    const int lane = t & 31;
    const long row = (long)blockIdx.x * 8 + wave;          // S*V rows total
    const float4 v = ((const float4*)(x + row * F_DIM))[lane];
    const float4 w = ((const float4*)Ws)[lane];
    float s = v.x * w.x + v.y * w.y + v.z * w.z + v.w * w.w;
    #pragma unroll
    for (int off = 16; off > 0; off >>= 1) s += __shfl_down(s, off, 32);
    if (lane == 0) support[row] = s;
}

// ---------------------------------------------------------------------------
// 4) score = support @ An + b    (f32 WMMA 16x16x4, one wave per 16x16 tile)
//    A frag (16x4): lanes 0-15 -> M=lane, half selects K pair {0,1}/{2,3}
//    B frag (4x16): lanes 0-15 -> N=lane, half selects K pair
//    C/D (16x16):   VGPR j, half h -> row = 8*h + j, col = lane&15
// ---------------------------------------------------------------------------
__global__ void score_wmma_kernel(const float* __restrict__ Amat,  // support [S,V]
                                  const float* __restrict__ Bmat,  // An [V,V]
                                  const float* __restrict__ bptr,
                                  float* __restrict__ C) {
    const int wave = threadIdx.x >> 5;                 // 4 waves / block
    const int lane = threadIdx.x & 31;
    const int tile = blockIdx.x * 4 + wave;            // 4096 tiles
    const int tilesN = V_DIM / 16;                     // 64
    const int tm = (tile / tilesN) * 16;
    const int tn = (tile % tilesN) * 16;
    const int hl = lane >> 4;                          // half-wave select
    const int l  = lane & 15;

    v8f acc = {};
    for (int k = 0; k < V_DIM; k += 4) {
        const int ka = k + hl * 2;
        v2f a, bfrag;
        a.x     = Amat[(tm + l) * V_DIM + ka];
        a.y     = Amat[(tm + l) * V_DIM + ka + 1];
        bfrag.x = Bmat[ka * V_DIM + tn + l];
        bfrag.y = Bmat[(ka + 1) * V_DIM + tn + l];
        acc = __builtin_amdgcn_wmma_f32_16x16x4_f32(
            /*neg_a=*/false, a, /*neg_b=*/false, bfrag,
            /*c_mod=*/(short)0, acc, /*reuse_a=*/false, /*reuse_b=*/false);
    }
    const float bias = bptr[0];
    #pragma unroll
    for (int j = 0; j < 8; ++j)
        C[(tm + hl * 8 + j) * V_DIM + tn + l] = acc[j] + bias;
}

// ---------------------------------------------------------------------------
// 5) gene_mean[u] = mean_s score[s,u]
// ---------------------------------------------------------------------------
__global__ void colmean_kernel(const float* __restrict__ score, float* __restrict__ mean) {
    const int u = blockIdx.x * blockDim.x + threadIdx.x;   // 1024 threads
    float s = 0.f;
    for (int r = 0; r < S_DIM; ++r) s += score[r * V_DIM + u];
    mean[u] = s * (1.0f / S_DIM);
}

// ---------------------------------------------------------------------------
// 6) deterministic top-K by rank (matches jax.lax.top_k ordering:
//    descending value, ties broken by smaller index)
// ---------------------------------------------------------------------------
__global__ void topk_kernel(const float* __restrict__ mean, int* __restrict__ idx) {
    __shared__ float sm[V_DIM];
    const int u = threadIdx.x;                             // 1024 threads, 1 block
    sm[u] = mean[u];
    __syncthreads();
    const float mu = sm[u];
    int rank = 0;
    for (int v = 0; v < V_DIM; ++v) {
        const float mv = sm[v];
        rank += (mv > mu) || (mv == mu && v < u);
    }
    if (rank < K_DIM) idx[rank] = u;
}

// ---------------------------------------------------------------------------
// 7) hidden[s,k,:] = x[s, idx[k], :] * tanh(score[s, idx[k]])
// ---------------------------------------------------------------------------
__global__ void gather_kernel(const float* __restrict__ x, const float* __restrict__ score,
                              const int* __restrict__ idx, float* __restrict__ hidden) {
    const int t = threadIdx.x;
    const int wave = t >> 5;
    const int lane = t & 31;
    const long row = (long)blockIdx.x * 8 + wave;          // S*K rows
    const int s = (int)(row >> 9);                         // / K_DIM
    const int k = (int)(row & (K_DIM - 1));
    const int g = idx[k];
    const float sc = tanhf(score[s * V_DIM + g]);
    float4 v = ((const float4*)(x + ((long)s * V_DIM + g) * F_DIM))[lane];
    v.x *= sc; v.y *= sc; v.z *= sc; v.w *= sc;
    ((float4*)(hidden + row * F_DIM))[lane] = v;
}

// ---------------------------------------------------------------------------
// 8) filtered degree: d2[i] = rsqrt(1 + sum_j adj[idx[i], idx[j]])
// ---------------------------------------------------------------------------
__global__ void fdeg_kernel(const float* __restrict__ adj, const int* __restrict__ idx,
                            float* __restrict__ d2) {
    __shared__ float red[256];
    const int gi = idx[blockIdx.x];                        // K blocks
    float s = 0.f;
    for (int j = threadIdx.x; j < K_DIM; j += 256) s += adj[gi * V_DIM + idx[j]];
    red[threadIdx.x] = s;
    __syncthreads();
    for (int off = 128; off > 0; off >>= 1) {
        if (threadIdx.x < off) red[threadIdx.x] += red[threadIdx.x + off];
        __syncthreads();
    }
    if (threadIdx.x == 0) d2[blockIdx.x] = rsqrtf(red[0] + 1.0f);
}

// ---------------------------------------------------------------------------
// 9) filtered_adj[i,j] = d2[i] * (adj[idx[i],idx[j]] + (i==j)) * d2[j]
// ---------------------------------------------------------------------------
__global__ void fadj_kernel(const float* __restrict__ adj, const int* __restrict__ idx,
                            const float* __restrict__ d2, float* __restrict__ out) {
    const int t = blockIdx.x * blockDim.x + threadIdx.x;   // K*K = 262144
    const int i = t >> 9;
    const int j = t & (K_DIM - 1);
    const float a = adj[(long)idx[i] * V_DIM + idx[j]] + (i == j ? 1.0f : 0.0f);
    out[t] = d2[i] * a * d2[j];
}

extern "C" void kernel_launch(void* const* d_in, const int* in_sizes, int n_in,
                              void* d_out, int out_size, void* d_ws, size_t ws_size,
                              hipStream_t stream) {
    const float* x   = (const float*)d_in[0];   // [S*V, F]
    const float* adj = (const float*)d_in[1];   // [V, V]
    const float* W   = (const float*)d_in[2];   // [F, 1]
    const float* b   = (const float*)d_in[3];   // [1]

    float* hidden = (float*)d_out;                                   // S*K*F
    float* fadj   = (float*)d_out + (long)S_DIM * K_DIM * F_DIM;     // K*K

    // workspace layout (floats): dinv | An | support | score | mean | d2 | idx
    float* ws      = (float*)d_ws;
    float* dinv    = ws;                                  // V
    float* An      = dinv + V_DIM;                        // V*V
    float* support = An + (long)V_DIM * V_DIM;            // S*V
    float* score   = support + (long)S_DIM * V_DIM;       // S*V
    float* mean    = score + (long)S_DIM * V_DIM;         // V
    float* d2      = mean + V_DIM;                        // K
    int*   idx     = (int*)(d2 + K_DIM);                  // K ints

    // A_norm
    deg_kernel<<<V_DIM, 256, 0, stream>>>(adj, dinv);
    an_kernel<<<(V_DIM * V_DIM) / 256, 256, 0, stream>>>(adj, dinv, An);

    // support = x @ W  (wave per row)
    support_kernel<<<(S_DIM * V_DIM) / 8, 256, 0, stream>>>(x, W, support);

    // score = support @ An + b  (WMMA f32 16x16x4)
    score_wmma_kernel<<<(S_DIM / 16) * (V_DIM / 16) / 4, 128, 0, stream>>>(support, An, b, score);

    // gene_mean -> deterministic top-K
    colmean_kernel<<<V_DIM / 256, 256, 0, stream>>>(score, mean);
    topk_kernel<<<1, V_DIM, 0, stream>>>(mean, idx);

    // hidden gather + tanh scale
    gather_kernel<<<(S_DIM * K_DIM) / 8, 256, 0, stream>>>(x, score, idx, hidden);

    // filtered adjacency renormalization
    fdeg_kernel<<<K_DIM, 256, 0, stream>>>(adj, idx, d2);
    fadj_kernel<<<(K_DIM * K_DIM) / 256, 256, 0, stream>>>(adj, idx, d2, fadj);
}